// OTAMMatching_88785563943235
// MI455X (gfx1250) — compile-verified
//
#include <hip/hip_runtime.h>

// OTAM cumulative distance for [400,400,16,16] similarity -> [400,400] votes.
// One lane per (q,s) tile. Both DP orientations fused into a single streaming
// pass over tile rows: forward DP row-major + transposed DP column-major.
// wave32 / gfx1250: native v_exp_f32 / v_log_f32 (base-2), global_prefetch_b8.

#define OTAM_L 16
#define OTAM_M 16
// 1/lambda * log2(e)  with lambda = 0.1
#define K_EXP 14.4269504088896340736f
// lambda * ln(2)
#define K_LOG 0.0693147180559945309f

#if __has_builtin(__builtin_amdgcn_exp2f)
#define EXP2F(x) __builtin_amdgcn_exp2f(x)
#else
#define EXP2F(x) exp2f(x)
#endif

#if __has_builtin(__builtin_amdgcn_logf)
#define LOG2F(x) __builtin_amdgcn_logf(x)   // v_log_f32 = log2
#else
#define LOG2F(x) __log2f(x)
#endif

// -lambda*log(exp(-a/l)+exp(-b/l)) = min(a,b) - l*ln(1+e^{-|a-b|/l})
__device__ __forceinline__ float softmin2(float a, float b) {
    float m = fminf(a, b);
    float d = fabsf(a - b) * K_EXP;
    float p = EXP2F(-d);
    return m - K_LOG * LOG2F(1.0f + p);
}

__global__ __launch_bounds__(256) void otam_kernel(const float* __restrict__ sim,
                                                   float* __restrict__ out,
                                                   int n_pairs) {
    int tid = blockIdx.x * blockDim.x + threadIdx.x;
    if (tid >= n_pairs) return;

    const float* tp = sim + (size_t)tid * (OTAM_L * OTAM_M);

    // Forward DP state: A[m] = C[l][m], m = 0..17 (A[0] is the implicit 0 col).
    float A[18];
    // Transposed DP state: B[l'] = C'[l'][m'-1], l' = 0..15.
    float B[16];
#pragma unroll
    for (int i = 0; i < 16; ++i) B[i] = 0.0f;
    A[0] = 0.0f;

    float row[16];

    for (int r = 0; r < OTAM_L; ++r) {
        // ---- load row r of the tile, convert to distance = 1 - sim ----
        const float4* rp = reinterpret_cast<const float4*>(tp) + r * 4;
        float4 q0 = rp[0], q1 = rp[1], q2 = rp[2], q3 = rp[3];
        row[0]  = 1.0f - q0.x; row[1]  = 1.0f - q0.y; row[2]  = 1.0f - q0.z; row[3]  = 1.0f - q0.w;
        row[4]  = 1.0f - q1.x; row[5]  = 1.0f - q1.y; row[6]  = 1.0f - q1.z; row[7]  = 1.0f - q1.w;
        row[8]  = 1.0f - q2.x; row[9]  = 1.0f - q2.y; row[10] = 1.0f - q2.z; row[11] = 1.0f - q2.w;
        row[12] = 1.0f - q3.x; row[13] = 1.0f - q3.y; row[14] = 1.0f - q3.z; row[15] = 1.0f - q3.w;

        if (r + 2 < OTAM_L) {
            // gfx1250 prefetch path (global_prefetch_b8); no counter impact.
            __builtin_prefetch((const void*)(tp + (r + 2) * OTAM_M), 0, 0);
        }

        // ---- forward DP: row r (row-major update, in place over A) ----
        if (r == 0) {
            float run = 0.0f;
#pragma unroll
            for (int m = 1; m <= 16; ++m) { run += row[m - 1]; A[m] = run; }
            A[17] = run;  // padded col: d = 0
        } else {
            float left = 0.0f;   // C[r][0]
            float diag = 0.0f;   // C[r-1][0]
#pragma unroll
            for (int m = 1; m <= 17; ++m) {
                float up = A[m];                 // C[r-1][m]
                float t = softmin2(diag, left);  // same order as reference LAE
                if (m == 1 || m == 17) t = softmin2(t, up);
                float P = (m <= 16) ? row[m - 1] : 0.0f;
                float val = P + t;
                A[m] = val;
                left = val;
                diag = up;
            }
        }

        // ---- transposed DP: column m' = r+1 (column-major update over B) ----
        {
            const bool bnd = (r == 0);  // m'==1 is a boundary column
            float diag = B[0];          // C'[l'-1][m'-1] for l'=1
            float nb = B[0] + row[0];   // row-0 prefix rule: C'[0][m'] = C'[0][m'-1] + d
            B[0] = nb;
            float prev_new = nb;        // C'[l'-1][m'] (boundary "up" term)
#pragma unroll
            for (int l = 1; l < 16; ++l) {
                float oldB = B[l];               // C'[l][m'-1]
                float t = softmin2(diag, oldB);
                if (bnd) t = softmin2(t, prev_new);
                float val = row[l] + t;
                diag = oldB;
                B[l] = val;
                prev_new = val;
            }
        }
    }

    // ---- transposed DP epilogue: padded boundary column m' = 17, d = 0 ----
    {
        float diag = B[0];
        float prev_new = B[0];  // C'[0][17] = C'[0][16] + 0
#pragma unroll
        for (int l = 1; l < 16; ++l) {
            float oldB = B[l];
            float t = softmin2(diag, oldB);
            t = softmin2(t, prev_new);  // boundary column
            diag = oldB;
            B[l] = t;
            prev_new = t;
        }
    }

    float fwd = A[17];
    float twd = B[15];
    out[tid] = -0.5f * (fwd + twd);
}

extern "C" void kernel_launch(void* const* d_in, const int* in_sizes, int n_in,
                              void* d_out, int out_size, void* d_ws, size_t ws_size,
                              hipStream_t stream) {
    (void)n_in; (void)d_ws; (void)ws_size; (void)out_size;
    const float* sim = (const float*)d_in[0];
    float* out = (float*)d_out;
    int n_pairs = in_sizes[0] / (OTAM_L * OTAM_M);  // 400*400 = 160000
    int block = 256;
    int grid = (n_pairs + block - 1) / block;
    otam_kernel<<<grid, block, 0, stream>>>(sim, out, n_pairs);
}